// MSSCA_84052509982729
// MI455X (gfx1250) — compile-verified
//
#include <hip/hip_runtime.h>
#include <float.h>

// ---------------- problem constants (match reference) ----------------
#define N_PTS   16384
#define C_IN    64
#define C_OUT   128
#define K_MAX   32
#define BN_EPS  1e-5f

typedef float v2f __attribute__((ext_vector_type(2)));
typedef float v8f __attribute__((ext_vector_type(8)));

// ---------------- workspace layout (in floats) ----------------
// [0, 256)                    BN accumulators: sum[128], sumsq[128]
// [256, 256+4*N)              p4 SoA: float4 {x, y, z, |p|^2}
// [256+4*N, +N*C_OUT)         h buffer (pre-BN, then post-BN in place)   (~8.25 MB total)
#define WS_BNS  0
#define WS_P4   256
#define WS_H    (256 + 4 * N_PTS)

// d_out layout (floats): [ p : 3*N ][ out : N*512 row-major ][ o : 1 ]
#define OUT_P      0
#define OUT_FEAT   (3 * N_PTS)
#define OUT_O      (3 * N_PTS + (size_t)N_PTS * 4 * C_OUT)

// =====================================================================
// Kernel A: zero BN accumulators, build float4 SoA, pass-through p / o.
// =====================================================================
__global__ void prep_kernel(const float* __restrict__ p,
                            float* __restrict__ ws,
                            float* __restrict__ outp) {
  int t = blockIdx.x * blockDim.x + threadIdx.x;
  if (t < 256) ws[WS_BNS + t] = 0.f;                 // must re-zero every call
  if (t < N_PTS) {
    float xx = p[3 * t + 0], yy = p[3 * t + 1], zz = p[3 * t + 2];
    float4* p4 = (float4*)(ws + WS_P4);
    p4[t] = make_float4(xx, yy, zz, xx * xx + yy * yy + zz * zz);
  }
  if (t < 3 * N_PTS) outp[OUT_P + t] = p[t];         // tuple elem 0: p
  if (t == 0) outp[OUT_O] = (float)N_PTS;            // tuple elem 2: o (=[N])
}

// =====================================================================
// Kernel B: h_pre = x @ W + b via V_WMMA_F32_16X16X4_F32, + BN partials.
// Block = 256 threads = 8 waves. Block owns 16 rows; wave w owns cols
// [16w, 16w+16). K loop: 64 in steps of 4 -> 16 WMMAs per wave (asm-
// confirmed: 16x v_wmma_f32_16x16x4_f32 with pipelined load clauses).
//
// f32 WMMA operand layout (ISA 7.12.2):
//   A 16x4 : lanes 0-15 hold {K=0,K=1} in v[0],v[1]; lanes 16-31 hold {K=2,K=3}
//   B 4x16 : lanes 0-15 hold rows {K=0,K=1}; lanes 16-31 rows {K=2,K=3}
//   C/D    : VGPR r -> row (lane<16 ? r : r+8), col = lane&15
// =====================================================================
__global__ void __launch_bounds__(256)
gemm_bn_kernel(const float* __restrict__ x, const float* __restrict__ W,
               const float* __restrict__ bias, float* __restrict__ ws) {
  float* hbuf = ws + WS_H;
  float* bns  = ws + WS_BNS;
  const int wave  = threadIdx.x >> 5;
  const int lane  = threadIdx.x & 31;
  const int m0    = blockIdx.x * 16;
  const int n0    = wave * 16;
  const int idx16 = lane & 15;          // A row / B col for this lane
  const int khalf = (lane >> 4) << 1;   // 0 or 2: which K pair this half-wave holds

  v8f acc = {0.f, 0.f, 0.f, 0.f, 0.f, 0.f, 0.f, 0.f};
#pragma unroll
  for (int k0 = 0; k0 < C_IN; k0 += 4) {
    // A tile: X[m0+idx16][k0+khalf .. +1]  (8B-aligned pair -> b64 load)
    float2 av = *(const float2*)(x + (m0 + idx16) * C_IN + k0 + khalf);
    v2f a; a.x = av.x; a.y = av.y;
    // B tile: W[k0+khalf][n0+idx16], W[k0+khalf+1][n0+idx16]
    v2f b;
    b.x = W[(k0 + khalf    ) * C_OUT + n0 + idx16];
    b.y = W[(k0 + khalf + 1) * C_OUT + n0 + idx16];
    acc = __builtin_amdgcn_wmma_f32_16x16x4_f32(false, a, false, b,
                                                (short)0, acc, false, false);
  }

  const int col = lane & 15;
  const int rb  = (lane >> 4) << 3;     // row base 0 or 8
  const float bv = bias[n0 + col];
  float s = 0.f, s2 = 0.f;
#pragma unroll
  for (int r = 0; r < 8; ++r) {
    float v = acc[r] + bv;
    hbuf[(size_t)(m0 + rb + r) * C_OUT + n0 + col] = v;
    s += v; s2 += v * v;
  }
  // lanes L and L^16 hold the same column -> combine, one atomic per column
  s  += __shfl_xor(s, 16, 32);
  s2 += __shfl_xor(s2, 16, 32);
  if (lane < 16) {
    atomicAdd(&bns[n0 + col], s);              // global_atomic_add_f32
    atomicAdd(&bns[C_OUT + n0 + col], s2);
  }
}

// =====================================================================
// Kernel C: BN finalize + ReLU (in place) + write h slice of output.
// =====================================================================
__global__ void bn_relu_kernel(const float* __restrict__ gamma,
                               const float* __restrict__ beta,
                               float* __restrict__ ws,
                               float* __restrict__ outp) {
  size_t t = (size_t)blockIdx.x * blockDim.x + threadIdx.x;   // < N*C_OUT
  int c = (int)(t & (C_OUT - 1));
  float* hbuf = ws + WS_H;
  const float* bns = ws + WS_BNS;
  float mu  = bns[c] * (1.f / N_PTS);
  float var = bns[C_OUT + c] * (1.f / N_PTS) - mu * mu;
  float inv = rsqrtf(var + BN_EPS) * gamma[c];
  float v = (hbuf[t] - mu) * inv + beta[c];
  v = fmaxf(v, 0.f);
  hbuf[t] = v;
  size_t row = t >> 7;
  outp[OUT_FEAT + row * (4 * C_OUT) + c] = v;
}

// =====================================================================
// Kernel D: exact KNN top-32 (sorted) + grouped prefix means.
// Block = 8 waves, one query row per wave. Candidate float4s streamed
// through a DOUBLE-BUFFERED 2x16KB LDS pipeline using
// GLOBAL_LOAD_ASYNC_TO_LDS_B128: tile t+1's copy (4 async ops/wave,
// VGPR-free, ASYNCcnt-tracked) flies underneath tile t's selection
// VALU work; async loads complete in order, so `s_wait_asynccnt 4`
// retires exactly the current tile while the next stays in flight.
// The wave's 32 lanes ARE the top-32 list (one (dist,idx) per lane);
// a wave-uniform threshold filters candidates, rare hits do a
// shfl-based argmax-replace. Final bitonic sort across lanes gives
// ascending distance (== reference top_k order).
// Score p2[j] - 2*pi.pj  (per-row constant p2[i] dropped: order-invariant).
// =====================================================================
#define TILE_PTS 1024
#define N_TILES  (N_PTS / TILE_PTS)

__device__ __forceinline__ void stage_tile_async(unsigned lds_base,
                                                 const float4* gsrc) {
  // 256 threads x 4 wave-wide b128 async ops = 16 KB; 4 ops per wave.
#pragma unroll
  for (int i = 0; i < TILE_PTS / 256; ++i) {
    int e = (int)threadIdx.x + i * 256;
    unsigned           laddr = lds_base + (unsigned)e * 16u;
    unsigned long long gaddr = (unsigned long long)(size_t)(gsrc + e);
    asm volatile("global_load_async_to_lds_b128 %0, %1, off"
                 :: "v"(laddr), "v"(gaddr) : "memory");
  }
}

__global__ void __launch_bounds__(256)
knn_group_kernel(const float* __restrict__ ws_c, float* __restrict__ outp) {
  const float4* p4   = (const float4*)(ws_c + WS_P4);
  const float*  hbuf = ws_c + WS_H;
  __shared__ float4 tile[2][TILE_PTS];              // 2 x 16 KB

  const int wave = threadIdx.x >> 5;
  const int lane = threadIdx.x & 31;
  const int row  = blockIdx.x * 8 + wave;

  const float4 q = p4[row];
  float bd = FLT_MAX;                               // this lane's slot value
  int   bj = 0;                                     // this lane's slot index
  float thr = FLT_MAX;                              // wave-uniform 32nd-best

  // LDS byte offsets of the two buffers = low 32 bits of their flat
  // addresses (ISA 10.2: LDS aperture keeps the offset in addr[31:0]).
  const unsigned lds0 = (unsigned)(size_t)(&tile[0][0]);
  const unsigned lds1 = (unsigned)(size_t)(&tile[1][0]);

  stage_tile_async(lds0, p4);                       // prologue: tile 0

  for (int t = 0; t < N_TILES; ++t) {
    if (t + 1 < N_TILES) {                          // prefetch tile t+1
      stage_tile_async((t & 1) ? lds0 : lds1, p4 + (t + 1) * TILE_PTS);
      asm volatile("s_wait_asynccnt 0x4" ::: "memory");  // tile t landed
    } else {
      asm volatile("s_wait_asynccnt 0x0" ::: "memory");  // last tile landed
    }
    __syncthreads();                                // tile t visible to block

    const float4* cur = tile[t & 1];
    const int jbase = t * TILE_PTS;
    for (int c0 = 0; c0 < TILE_PTS; c0 += 32) {
      float4 cp = cur[c0 + lane];
      float d = cp.w - 2.f * (q.x * cp.x + q.y * cp.y + q.z * cp.z);
      int   j = jbase + c0 + lane;
      unsigned long long m = __ballot(d < thr);
      while (m) {
        int src = __ffsll(m) - 1;
        m &= m - 1;
        float dc = __shfl(d, src, 32);
        int   jc = __shfl(j, src, 32);
        if (dc < thr) {                             // re-check vs updated thr
          // wave argmax over current slots -> lane holding the worst entry
          float mx = bd; int ml = lane;
#pragma unroll
          for (int off = 16; off > 0; off >>= 1) {
            float om = __shfl_xor(mx, off, 32);
            int   ol = __shfl_xor(ml, off, 32);
            if (om > mx) { mx = om; ml = ol; }
          }
          if (lane == ml) { bd = dc; bj = jc; }     // replace worst
          float nm = bd;                            // new threshold = new max
#pragma unroll
          for (int off = 16; off > 0; off >>= 1) {
            float om = __shfl_xor(nm, off, 32);
            nm = om > nm ? om : nm;
          }
          thr = nm;
        }
      }
    }
    __syncthreads();                                // buffer free for reuse
  }

  // ---- bitonic sort (dist,idx) ascending across the 32 lanes ----
#pragma unroll
  for (int size = 2; size <= 32; size <<= 1) {
#pragma unroll
    for (int stride = size >> 1; stride > 0; stride >>= 1) {
      float od = __shfl_xor(bd, stride, 32);
      int   oj = __shfl_xor(bj, stride, 32);
      bool up      = ((lane & size) == 0);          // size==32: always true
      bool lower   = ((lane & stride) == 0);
      bool takeMin = (up == lower);
      bool repl    = takeMin ? (od < bd) : (od > bd);
      if (repl) { bd = od; bj = oj; }
    }
  }

  // ---- gather h rows (coalesced b128/lane) + prefix means at 8/16/32 ----
  float4 a  = make_float4(0.f, 0.f, 0.f, 0.f);
  float4 m8 = a, m16 = a;
  for (int k = 0; k < K_MAX; ++k) {
    int jk = __shfl(bj, k, 32);                     // broadcast k-th neighbor
    float4 hv = ((const float4*)(hbuf + (size_t)jk * C_OUT))[lane];
    a.x += hv.x; a.y += hv.y; a.z += hv.z; a.w += hv.w;
    if (k == 7)  m8  = a;
    if (k == 15) m16 = a;
  }
  float* ob = outp + OUT_FEAT + (size_t)row * (4 * C_OUT);
  ((float4*)(ob + 1 * C_OUT))[lane] =
      make_float4(m8.x * 0.125f,  m8.y * 0.125f,  m8.z * 0.125f,  m8.w * 0.125f);
  ((float4*)(ob + 2 * C_OUT))[lane] =
      make_float4(m16.x * 0.0625f, m16.y * 0.0625f, m16.z * 0.0625f, m16.w * 0.0625f);
  ((float4*)(ob + 3 * C_OUT))[lane] =
      make_float4(a.x * 0.03125f, a.y * 0.03125f, a.z * 0.03125f, a.w * 0.03125f);
}

// =====================================================================
extern "C" void kernel_launch(void* const* d_in, const int* in_sizes, int n_in,
                              void* d_out, int out_size, void* d_ws, size_t ws_size,
                              hipStream_t stream) {
  const float* p     = (const float*)d_in[0];
  const float* x     = (const float*)d_in[1];
  // d_in[2] = o (int32, single batch) — value known statically, unused on device
  const float* W     = (const float*)d_in[3];
  const float* b     = (const float*)d_in[4];
  const float* gamma = (const float*)d_in[5];
  const float* beta  = (const float*)d_in[6];
  float* outp = (float*)d_out;
  float* ws   = (float*)d_ws;

  prep_kernel   <<<(3 * N_PTS + 255) / 256, 256, 0, stream>>>(p, ws, outp);
  gemm_bn_kernel<<<N_PTS / 16,              256, 0, stream>>>(x, W, b, ws);
  bn_relu_kernel<<<(N_PTS * C_OUT) / 256,   256, 0, stream>>>(gamma, beta, ws, outp);
  knn_group_kernel<<<N_PTS / 8,             256, 0, stream>>>(ws, outp);
}